// InfoNCELoss_7103875907856
// MI455X (gfx1250) — compile-verified
//
#include <hip/hip_runtime.h>
#include <hip/hip_bf16.h>
#include <stdint.h>

typedef __attribute__((ext_vector_type(2))) float v2f;
typedef __attribute__((ext_vector_type(8))) float v8f;

#define B_DIM 256
#define T_DIM 128
#define V_DIM 512
#define S_DIM 256
#define TC    16   // t-chunk size staged in LDS

__device__ __forceinline__ float clip30(float v) {
    return fminf(fmaxf(v, -30.0f), 30.0f);
}

// -----------------------------------------------------------------------------
// Kernel A: terms[s,b] = sum_t clip(x[b,t,tgt[s,t]])  -> stored b-major W[b][s]
// One block per b, 256 threads = 8 waves; wave w owns s in [32w, 32w+32) as two
// 16x16 f32 WMMA accumulators (all-ones B matrix => D[m,*] += sum_k A[m,k]).
// x rows are staged global->LDS with async b128 copies, DOUBLE-BUFFERED so the
// DMA for chunk n+1 overlaps the gather+WMMA compute of chunk n (ASYNCcnt
// completes in order, so s_wait_asynccnt 8 retires exactly the older chunk).
// -----------------------------------------------------------------------------
__global__ __launch_bounds__(256) void terms_kernel(const float* __restrict__ x,
                                                    const int*   __restrict__ tgt,
                                                    float*       __restrict__ W) {
    __shared__ __align__(16) float xs[2][TC * V_DIM];  // 2 x 32 KB
    __shared__ __align__(16) int   ts[2][TC * S_DIM];  // 2 x 16 KB, [tl][s]

    const int b    = blockIdx.x;
    const int tid  = threadIdx.x;
    const int lane = tid & 31;
    const int wave = tid >> 5;
    const int sbase = wave * 32;          // this wave's first s
    const int M    = lane & 15;           // row within the 16x16 tile
    const int half = lane >> 4;           // A layout: lanes 16-31 hold K={2,3}

    v8f cA = {};                          // terms for s in [sbase, sbase+16)
    v8f cB = {};                          // terms for s in [sbase+16, sbase+32)
    const v2f ones = {1.0f, 1.0f};        // all-ones 4x16 B matrix

    const float* xb = x + (size_t)b * T_DIM * V_DIM;

    auto issue_chunk = [&](int tc, int buf) {
        // 8 async b128 loads per thread: 8192 f32 chunk -> xs[buf]
        const int      base = tid * 32;
        const uint32_t ldsb = (uint32_t)(uintptr_t)(&xs[buf][0]);
        #pragma unroll
        for (int i = 0; i < 8; ++i) {
            const int      eo = base + i * 4;
            const uint32_t la = ldsb + (uint32_t)(eo * 4);
            const uint64_t ga = (uint64_t)(uintptr_t)(xb + tc * TC * V_DIM + eo);
            asm volatile("global_load_async_to_lds_b128 %0, %1, off"
                         :: "v"(la), "v"(ga) : "memory");
        }
        // stage tgt chunk transposed: ts[buf][tl*256 + s] = tgt[s, tc*16+tl]
        const int4* tq = (const int4*)(tgt + tid * T_DIM + tc * TC);
        #pragma unroll
        for (int j = 0; j < 4; ++j) {
            int4 q = tq[j];
            ts[buf][(j * 4 + 0) * S_DIM + tid] = q.x;
            ts[buf][(j * 4 + 1) * S_DIM + tid] = q.y;
            ts[buf][(j * 4 + 2) * S_DIM + tid] = q.z;
            ts[buf][(j * 4 + 3) * S_DIM + tid] = q.w;
        }
    };

    auto compute_chunk = [&](int buf) {
        const float* xc = &xs[buf][0];
        const int*   tl = &ts[buf][0];
        #pragma unroll
        for (int kt = 0; kt < 4; ++kt) {
            const int tl0 = kt * 4 + half * 2;  // A VGPR0 holds K={0,2}
            const int tl1 = tl0 + 1;            // A VGPR1 holds K={1,3}
            {   // tile A: s = sbase + M
                const int s = sbase + M;
                v2f a;
                a.x = clip30(xc[tl0 * V_DIM + tl[tl0 * S_DIM + s]]);
                a.y = clip30(xc[tl1 * V_DIM + tl[tl1 * S_DIM + s]]);
                cA = __builtin_amdgcn_wmma_f32_16x16x4_f32(
                        false, a, false, ones, (short)0, cA, false, false);
            }
            {   // tile B: s = sbase + 16 + M
                const int s = sbase + 16 + M;
                v2f a;
                a.x = clip30(xc[tl0 * V_DIM + tl[tl0 * S_DIM + s]]);
                a.y = clip30(xc[tl1 * V_DIM + tl[tl1 * S_DIM + s]]);
                cB = __builtin_amdgcn_wmma_f32_16x16x4_f32(
                        false, a, false, ones, (short)0, cB, false, false);
            }
        }
    };

    // ---- software pipeline over 8 chunks ----
    issue_chunk(0, 0);
    #pragma unroll 1
    for (int tc = 0; tc < T_DIM / TC - 1; ++tc) {
        const int cur = tc & 1;
        issue_chunk(tc + 1, cur ^ 1);                  // prefetch next chunk
        asm volatile("s_wait_asynccnt 0x8" ::: "memory");  // older 8 retired
        __syncthreads();
        compute_chunk(cur);
        __syncthreads();                               // before buf reuse
    }
    asm volatile("s_wait_asynccnt 0x0" ::: "memory");
    __syncthreads();
    compute_chunk((T_DIM / TC - 1) & 1);

    // ---- extract terms: column N=0 of C/D lives in lanes 0 and 16 ----
    float* Wb = W + (size_t)b * S_DIM;
    if (lane == 0) {
        #pragma unroll
        for (int j = 0; j < 8; ++j) {
            Wb[sbase + j]      = cA[j];   // M = j
            Wb[sbase + 16 + j] = cB[j];
        }
    }
    if (lane == 16) {
        #pragma unroll
        for (int j = 0; j < 8; ++j) {
            Wb[sbase + 8 + j]  = cA[j];   // M = 8 + j
            Wb[sbase + 24 + j] = cB[j];
        }
    }
}

// -----------------------------------------------------------------------------
// Kernel B: per-s logsumexp over b, coef[s,b] = 1(b==s) - softmax, per-s loss.
// One block per s, thread = b. coef stored b-major C[b][s] so the scatter
// kernel reads it coalesced.
// -----------------------------------------------------------------------------
__global__ __launch_bounds__(256) void softmax_kernel(const float* __restrict__ W,
                                                      float* __restrict__ C,
                                                      float* __restrict__ losses) {
    __shared__ float red[S_DIM];
    const int s = blockIdx.x;
    const int b = threadIdx.x;

    const float val = W[(size_t)b * S_DIM + s];   // terms[s,b]

    red[b] = val;
    __syncthreads();
    #pragma unroll
    for (int off = 128; off > 0; off >>= 1) {
        if (b < off) red[b] = fmaxf(red[b], red[b + off]);
        __syncthreads();
    }
    const float mx = red[0];
    __syncthreads();

    const float e = __expf(val - mx);
    red[b] = e;
    __syncthreads();
    #pragma unroll
    for (int off = 128; off > 0; off >>= 1) {
        if (b < off) red[b] += red[b + off];
        __syncthreads();
    }
    const float sum = red[0];

    const float coef = ((b == s) ? 1.0f : 0.0f) - e / sum;
    C[(size_t)b * S_DIM + s] = coef;

    if (b == s) losses[s] = val - (mx + __logf(sum));
}

// -----------------------------------------------------------------------------
// Kernel B2: total_loss = sum_s losses[s] -> d_out[0]
// -----------------------------------------------------------------------------
__global__ __launch_bounds__(256) void loss_reduce(const float* __restrict__ losses,
                                                   float* __restrict__ out) {
    __shared__ float red[S_DIM];
    const int t = threadIdx.x;
    red[t] = losses[t];
    __syncthreads();
    #pragma unroll
    for (int off = 128; off > 0; off >>= 1) {
        if (t < off) red[t] += red[t + off];
        __syncthreads();
    }
    if (t == 0) out[0] = red[0];
}

// -----------------------------------------------------------------------------
// Kernel C: grad[b,t,v] = sum_s coef[s,b] * 1(tgt[s,t]==v)
// Block = (b, 16-t chunk). Rows built in LDS with ds_add_f32 atomics, then
// streamed out. grad base = d_out+1 is 4B-aligned, so each block's 8192-elem
// region starts at byte%16 == 4: elements [3, 8191) are stored as 2047 aligned
// b128s, plus 4 scalar edge elements.
// -----------------------------------------------------------------------------
__global__ __launch_bounds__(256) void grad_kernel(const int*   __restrict__ tgt,
                                                   const float* __restrict__ C,
                                                   float*       __restrict__ grad) {
    __shared__ __align__(16) float rows[TC * V_DIM];  // 32 KB
    __shared__ __align__(16) int   ts[TC * S_DIM];    // 16 KB, [tl][s]

    const int blk = blockIdx.x;
    const int b   = blk >> 3;           // T_DIM/TC == 8 chunks per b
    const int tc  = blk & 7;
    const int tid = threadIdx.x;        // == s for the scatter phase

    const float cf = C[(size_t)b * S_DIM + tid];   // coef[s=tid, b], coalesced

    // zero the 16 LDS rows (float4)
    float4* r4 = (float4*)rows;
    const float4 z = make_float4(0.f, 0.f, 0.f, 0.f);
    #pragma unroll
    for (int i = 0; i < 8; ++i) r4[tid + i * 256] = z;

    // stage tgt chunk transposed: ts[tl*256 + s] = tgt[s, tc*16+tl]
    {
        const int4* tq = (const int4*)(tgt + tid * T_DIM + tc * TC);
        #pragma unroll
        for (int j = 0; j < 4; ++j) {
            int4 q = tq[j];
            ts[(j * 4 + 0) * S_DIM + tid] = q.x;
            ts[(j * 4 + 1) * S_DIM + tid] = q.y;
            ts[(j * 4 + 2) * S_DIM + tid] = q.z;
            ts[(j * 4 + 3) * S_DIM + tid] = q.w;
        }
    }
    __syncthreads();

    // scatter-add: rows[tl][tgt[s, t]] += coef[s, b]   (ds_add_f32)
    #pragma unroll
    for (int j = 0; j < TC; ++j) {
        const int v = ts[j * S_DIM + tid];
        atomicAdd(&rows[j * V_DIM + v], cf);
    }
    __syncthreads();

    // stream rows to global: realigned b128 interior + scalar edges
    float* g = grad + ((size_t)b * T_DIM + tc * TC) * V_DIM;
    if (tid < 3)   g[tid]  = rows[tid];          // elements 0,1,2
    if (tid == 3)  g[8191] = rows[8191];         // last element
    #pragma unroll
    for (int i = 0; i < 8; ++i) {
        const int j = tid + i * 256;             // float4 index, [0, 2047)
        if (j < 2047) {
            float4 vv;
            vv.x = rows[3 + 4 * j + 0];
            vv.y = rows[3 + 4 * j + 1];
            vv.z = rows[3 + 4 * j + 2];
            vv.w = rows[3 + 4 * j + 3];
            *(float4*)(g + 3 + 4 * j) = vv;      // 16B-aligned b128 store
        }
    }
}

// -----------------------------------------------------------------------------
extern "C" void kernel_launch(void* const* d_in, const int* in_sizes, int n_in,
                              void* d_out, int out_size, void* d_ws, size_t ws_size,
                              hipStream_t stream) {
    const float* x   = (const float*)d_in[0];
    const int*   tgt = (const int*)d_in[1];
    // d_in[2] (sup_batchsize) == 256 == B_DIM, baked into the kernels.

    float* out = (float*)d_out;          // [0] = total_loss, [1..] = grad
    float* W      = (float*)d_ws;        // terms, b-major  [B][S]  (256 KB)
    float* C      = W + B_DIM * S_DIM;   // coef,  b-major  [B][S]  (256 KB)
    float* losses = C + B_DIM * S_DIM;   // per-s losses            (1 KB)

    terms_kernel  <<<B_DIM, 256, 0, stream>>>(x, tgt, W);
    softmax_kernel<<<S_DIM, 256, 0, stream>>>(W, C, losses);
    loss_reduce   <<<1,     256, 0, stream>>>(losses, out);
    grad_kernel   <<<B_DIM * (T_DIM / TC), 256, 0, stream>>>(tgt, C, out + 1);
}